// BlenderLayer_58746562674779
// MI455X (gfx1250) — compile-verified
//
#include <hip/hip_runtime.h>
#include <math.h>

#define GAUSS 128
#define TILE_P 16
#define GTILES (GAUSS / 16)

typedef __attribute__((ext_vector_type(2))) float v2f;
typedef __attribute__((ext_vector_type(8))) float v8f;

// ---------------------------------------------------------------------------
// Kernel 1: sort 128 gaussians by distance-to-camera, project to 2D, and emit
// per-gaussian packed data (sorted order):
//   [0..5] = k * bilinear-form features (k = -0.5*log2(e), so q' = F.H gives
//            exp2(q') = exp(-0.5 q)),  [6..7] = 0 (K padded to 8),
//   [8] = opacity w = sp/(1+sp),  [9..11] = squashed rgb.
// ---------------------------------------------------------------------------
__global__ __launch_bounds__(GAUSS) void splat_prep(
    const float* __restrict__ mu, const float* __restrict__ chol,
    const float* __restrict__ alpha, const float* __restrict__ rgb,
    const float* __restrict__ R, const float* __restrict__ tr,
    const float* __restrict__ Pr, float* __restrict__ gout)
{
  __shared__ float d2s[GAUSS];
  const int g = threadIdx.x;

  const float dx = mu[3*g+0] - tr[0];
  const float dy = mu[3*g+1] - tr[1];
  const float dz = mu[3*g+2] - tr[2];
  const float d2 = dx*dx + dy*dy + dz*dz;        // monotonic in norm
  d2s[g] = d2;
  __syncthreads();

  int rank = 0;
  for (int j = 0; j < GAUSS; ++j) {
    const float dj = d2s[j];
    rank += (dj < d2) || (dj == d2 && j < g);    // stable argsort rank
  }

  // Sigma = L L^T,  L = tril(chol) + 0.3 I
  const float L00 = chol[9*g+0] + 0.3f;
  const float L10 = chol[9*g+3];
  const float L11 = chol[9*g+4] + 0.3f;
  const float L20 = chol[9*g+6];
  const float L21 = chol[9*g+7];
  const float L22 = chol[9*g+8] + 0.3f;
  const float S00 = L00*L00;
  const float S01 = L00*L10;
  const float S02 = L00*L20;
  const float S11 = L10*L10 + L11*L11;
  const float S12 = L10*L20 + L11*L21;
  const float S22 = L20*L20 + L21*L21 + L22*L22;

  // mu_cam = R^T (mu - t)   (R row-major)
  const float mc0 = R[0]*dx + R[3]*dy + R[6]*dz;
  const float mc1 = R[1]*dx + R[4]*dy + R[7]*dz;
  const float mc2 = R[2]*dx + R[5]*dy + R[8]*dz;
  const float mx  = Pr[0]*mc0 + Pr[1]*mc1 + Pr[2]*mc2;
  const float my  = Pr[3]*mc0 + Pr[4]*mc1 + Pr[5]*mc2;

  // P_cam = Pr @ R^T  :  P_cam[i][k] = sum_j Pr[i][j] * R[k][j]
  const float P00 = Pr[0]*R[0] + Pr[1]*R[1] + Pr[2]*R[2];
  const float P01 = Pr[0]*R[3] + Pr[1]*R[4] + Pr[2]*R[5];
  const float P02 = Pr[0]*R[6] + Pr[1]*R[7] + Pr[2]*R[8];
  const float P10 = Pr[3]*R[0] + Pr[4]*R[1] + Pr[5]*R[2];
  const float P11 = Pr[3]*R[3] + Pr[4]*R[4] + Pr[5]*R[5];
  const float P12 = Pr[3]*R[6] + Pr[4]*R[7] + Pr[5]*R[8];

  // S2 = P_cam Sigma P_cam^T + 1e-4 I  (2x2 symmetric)
  const float M00 = P00*S00 + P01*S01 + P02*S02;
  const float M01 = P00*S01 + P01*S11 + P02*S12;
  const float M02 = P00*S02 + P01*S12 + P02*S22;
  const float M10 = P10*S00 + P11*S01 + P12*S02;
  const float M11 = P10*S01 + P11*S11 + P12*S12;
  const float M12 = P10*S02 + P11*S12 + P12*S22;
  const float A2 = M00*P00 + M01*P01 + M02*P02 + 1e-4f;
  const float Bc = M00*P10 + M01*P11 + M02*P12;
  const float C2 = M10*P10 + M11*P11 + M12*P12 + 1e-4f;

  const float inv_det = 1.0f / (A2*C2 - Bc*Bc);
  const float ia =  C2 * inv_det;     // q = ia dx^2 + 2 ib dx dy + ic dy^2
  const float ib = -Bc * inv_det;
  const float ic =  A2 * inv_det;

  const float kf = -0.72134752044448170368f;   // -0.5 * log2(e)
  const float F0 = kf * ia;
  const float F1 = kf * 2.0f * ib;
  const float F2 = kf * ic;
  const float F3 = kf * -2.0f * (ia*mx + ib*my);
  const float F4 = kf * -2.0f * (ib*mx + ic*my);
  const float F5 = kf * (ia*mx*mx + 2.0f*ib*mx*my + ic*my*my);

  const float av = alpha[g];
  const float sp = (av > 0.0f) ? (av + log1pf(expf(-av))) : log1pf(expf(av));
  const float w  = sp / (1.0f + sp);

  float* o = gout + rank * 12;
  o[0] = F0; o[1] = F1; o[2] = F2; o[3] = F3; o[4] = F4; o[5] = F5;
  o[6] = 0.0f; o[7] = 0.0f;
  o[8] = w;
  o[9]  = rgb[3*g+0] / (1.0f + fabsf(rgb[3*g+0]));
  o[10] = rgb[3*g+1] / (1.0f + fabsf(rgb[3*g+1]));
  o[11] = rgb[3*g+2] / (1.0f + fabsf(rgb[3*g+2]));
}

// ---------------------------------------------------------------------------
// Kernel 2: per wave, one 16-pixel tile. For each of 8 gaussian tiles:
// q' tile (16g x 16p) via 2x V_WMMA_F32_16X16X4_F32 (K=6 padded to 8), then
// per-lane sequential blend of its 8 gaussian rows, shfl_xor(16) segment
// combine, and fold into running (C, T). EXEC is all-ones around the WMMAs.
// ---------------------------------------------------------------------------
__global__ __launch_bounds__(256) void splat_main(
    const float* __restrict__ x, const float* __restrict__ gdata,
    const float* __restrict__ bg, float* __restrict__ out)
{
  __shared__ __align__(16) float sg[GAUSS * 12];
  for (int i = threadIdx.x; i < GAUSS * 12; i += 256) sg[i] = gdata[i];
  __syncthreads();

  const int lane = threadIdx.x & 31;
  const int wave = threadIdx.x >> 5;
  const int tile = blockIdx.x * 8 + wave;   // pixel tile index
  const int pcol = lane & 15;               // pixel column in tile (N)
  const int hi   = lane >> 4;               // which half-wave (M/K split)
  const int pix  = tile * TILE_P + pcol;    // flat pixel in [0, B*N)

  const float px = x[2*pix + 0];
  const float py = x[2*pix + 1];

  // B operand: H = [px^2, px*py, py^2, px, py, 1, 0, 0]
  // layout: vgpr0 = K(0|2), vgpr1 = K(1|3); second wmma: K(4|6), K(5|7)
  v2f Bm0, Bm1;
  Bm0.x = hi ? py*py : px*px;
  Bm0.y = hi ? px    : px*py;
  Bm1.x = hi ? 0.0f  : py;
  Bm1.y = hi ? 0.0f  : 1.0f;

  float Rr = 0.0f, Rg = 0.0f, Rb = 0.0f, TT = 1.0f;  // running composite

  for (int t = 0; t < GTILES; ++t) {
    // A operand: row M = pcol, features K(0|2),K(1|3) then K(4|6),K(5|7)
    const float* fa = &sg[(t*16 + pcol) * 12 + hi*2];
    v2f Am0, Am1;
    Am0.x = fa[0]; Am0.y = fa[1];
    Am1.x = fa[4]; Am1.y = fa[5];

    v8f D = {0.f, 0.f, 0.f, 0.f, 0.f, 0.f, 0.f, 0.f};
    D = __builtin_amdgcn_wmma_f32_16x16x4_f32(false, Am0, false, Bm0,
                                              (short)0, D, false, false);
    D = __builtin_amdgcn_wmma_f32_16x16x4_f32(false, Am1, false, Bm1,
                                              (short)0, D, false, false);

    // this lane holds q' for gaussian rows (hi*8 + j), j = 0..7, pixel pcol
    float Cr = 0.0f, Cg = 0.0f, Cb = 0.0f, Tt = 1.0f;
    const int gbase = t*16 + hi*8;
#pragma unroll
    for (int j = 0; j < 8; ++j) {
      const float gv = __builtin_amdgcn_exp2f(D[j]);           // v_exp_f32
      const float4 wc = *(const float4*)&sg[(gbase + j)*12 + 8]; // w,r,g,b
      const float a  = wc.x * gv;
      const float om = 1.0f - a;
      Cr = Cr*om + wc.y*a;
      Cg = Cg*om + wc.z*a;
      Cb = Cb*om + wc.w*a;
      Tt *= om;
    }

    // combine halves: lo lane = rows 0..7 (earlier), hi lane = rows 8..15
    const float oCr = __shfl_xor(Cr, 16, 32);
    const float oCg = __shfl_xor(Cg, 16, 32);
    const float oCb = __shfl_xor(Cb, 16, 32);
    const float oT  = __shfl_xor(Tt, 16, 32);
    const float Th   = hi ? Tt : oT;                 // T of the later half
    const float C16r = (hi ? oCr : Cr) * Th + (hi ? Cr : oCr);
    const float C16g = (hi ? oCg : Cg) * Th + (hi ? Cg : oCg);
    const float C16b = (hi ? oCb : Cb) * Th + (hi ? Cb : oCb);
    const float T16  = Tt * oT;

    // fold tile (later gaussians) into running composite
    Rr = Rr*T16 + C16r;
    Rg = Rg*T16 + C16g;
    Rb = Rb*T16 + C16b;
    TT *= T16;
  }

  if (hi == 0) {
    float* o = out + pix*3;
    o[0] = Rr + TT*bg[0];
    o[1] = Rg + TT*bg[1];
    o[2] = Rb + TT*bg[2];
  }
}

// ---------------------------------------------------------------------------
extern "C" void kernel_launch(void* const* d_in, const int* in_sizes, int n_in,
                              void* d_out, int out_size, void* d_ws, size_t ws_size,
                              hipStream_t stream)
{
  const float* x     = (const float*)d_in[0];
  const float* mu    = (const float*)d_in[1];
  const float* chol  = (const float*)d_in[2];
  const float* alpha = (const float*)d_in[3];
  const float* rgb   = (const float*)d_in[4];
  const float* rot   = (const float*)d_in[5];
  const float* trl   = (const float*)d_in[6];
  const float* proj  = (const float*)d_in[7];
  const float* bg    = (const float*)d_in[8];
  float* out   = (float*)d_out;
  float* gdata = (float*)d_ws;   // 128 * 12 floats = 6 KB

  splat_prep<<<1, GAUSS, 0, stream>>>(mu, chol, alpha, rgb, rot, trl, proj, gdata);

  const int P     = in_sizes[0] / 2;   // B*N pixels
  const int tiles = P / TILE_P;        // 16384
  splat_main<<<tiles / 8, 256, 0, stream>>>(x, gdata, bg, out);
}